// LGSSM_37984690766376
// MI455X (gfx1250) — compile-verified
//
#include <hip/hip_runtime.h>
#include <hip/hip_bf16.h>

typedef float v2f __attribute__((ext_vector_type(2)));
typedef float v8f __attribute__((ext_vector_type(8)));

constexpr int BATCH = 256, TLEN = 256, DZ = 32, DA = 16, DU = 8, KMIX = 4, HDIM = 64;

// Output tuple layout: (z, mu, Sig, a, A_t, B_t, C_t), each (BATCH, TLEN, ...)
constexpr size_t OFF_Z   = 0;
constexpr size_t OFF_MU  = OFF_Z   + (size_t)BATCH * TLEN * DZ;        //   2,097,152
constexpr size_t OFF_SIG = OFF_MU  + (size_t)BATCH * TLEN * DZ;        //   4,194,304
constexpr size_t OFF_A   = OFF_SIG + (size_t)BATCH * TLEN * DZ * DZ;   //  71,303,168
constexpr size_t OFF_AT  = OFF_A   + (size_t)BATCH * TLEN * DA;        //  72,351,744
constexpr size_t OFF_BT  = OFF_AT  + (size_t)BATCH * TLEN * DZ * DZ;   // 139,460,608
constexpr size_t OFF_CT  = OFF_BT  + (size_t)BATCH * TLEN * DZ * DU;   // 156,237,824

__device__ __forceinline__ float sigm(float x) { return 1.0f / (1.0f + __expf(-x)); }

// One wave (32 lanes) per batch element; sequential rollout over 256 steps.
__global__ __launch_bounds__(32) void lgssm_rollout(
    const float* __restrict__ mu0,   const float* __restrict__ Sig0,
    const float* __restrict__ alpha0,const float* __restrict__ hinit,
    const float* __restrict__ uf,    const float* __restrict__ epsg,
    const float* __restrict__ Ag,    const float* __restrict__ Bg,
    const float* __restrict__ Cg,    const float* __restrict__ Qg,
    const float* __restrict__ Wx,    const float* __restrict__ Wh,
    const float* __restrict__ bxp,   const float* __restrict__ bhp,
    const float* __restrict__ Wo,    const float* __restrict__ bop,
    float* __restrict__ out)
{
    // Staged constants (t-invariant, reused 256x)
    __shared__ float sAg[KMIX * DZ * DZ];   // 16 KB
    __shared__ float sBg[KMIX * DZ * DU];   //  4 KB
    __shared__ float sCg[KMIX * DA * DZ];   //  8 KB
    // Working tiles, stride-33 padding (odd stride -> conflict-free columns)
    __shared__ float sAt[DZ * 33], sS[DZ * 33], sT[DZ * 33], sQ[DZ * 33];
    __shared__ float sBt[DZ * 9];
    __shared__ float sCt[DA * 33];
    __shared__ float zv[DZ], hv[HDIM], uv[DU], salpha[KMIX];

    const int lane = threadIdx.x;
    const int b    = blockIdx.x;
    const int lo   = lane & 15;      // WMMA N / M index within 16
    const int hi4  = lane >> 4;      // WMMA K-half selector

    // ---- one-time staging ----
#pragma unroll 4
    for (int i = 0; i < KMIX * DZ; ++i) sAg[i * 32 + lane] = Ag[i * 32 + lane];
#pragma unroll
    for (int i = 0; i < KMIX * DU; ++i) sBg[i * 32 + lane] = Bg[i * 32 + lane];
#pragma unroll
    for (int i = 0; i < KMIX * DA; ++i) sCg[i * 32 + lane] = Cg[i * 32 + lane];
#pragma unroll
    for (int i = 0; i < DZ; ++i) sQ[i * 33 + lane] = Qg[i * 32 + lane];

    float mur = mu0[b * DZ + lane];                         // lane i holds mu_i
#pragma unroll
    for (int i = 0; i < DZ; ++i)
        sS[i * 33 + lane] = Sig0[(size_t)b * DZ * DZ + i * 32 + lane];
    if (lane < KMIX) salpha[lane] = alpha0[b * KMIX + lane];
    float hr0 = hinit[b * HDIM + lane];
    float hr1 = hinit[b * HDIM + 32 + lane];
    hv[lane] = hr0; hv[lane + 32] = hr1;
    __syncthreads();

    for (int t = 0; t < TLEN; ++t) {
        const size_t bt = (size_t)b * TLEN + t;
        if (lane < DU) uv[lane] = uf[bt * DU + lane];
        const float epsr = epsg[bt * DZ + lane];
        const float al0 = salpha[0], al1 = salpha[1], al2 = salpha[2], al3 = salpha[3];

        // ---- alpha-weighted mixtures (store to LDS + streaming global) ----
        {
            float* oA = out + OFF_AT + bt * (DZ * DZ);
#pragma unroll 4
            for (int i = 0; i < 32; ++i) {
                const int e = i * 32 + lane;
                float v = al0 * sAg[e] + al1 * sAg[1024 + e] + al2 * sAg[2048 + e] + al3 * sAg[3072 + e];
                sAt[i * 33 + lane] = v;
                __builtin_nontemporal_store(v, oA + e);
            }
        }
        {
            float* oB = out + OFF_BT + bt * (DZ * DU);
#pragma unroll
            for (int i = 0; i < 8; ++i) {
                const int e = i * 32 + lane;
                float v = al0 * sBg[e] + al1 * sBg[256 + e] + al2 * sBg[512 + e] + al3 * sBg[768 + e];
                sBt[(e >> 3) * 9 + (e & 7)] = v;
                __builtin_nontemporal_store(v, oB + e);
            }
        }
        {
            float* oC = out + OFF_CT + bt * (DA * DZ);
#pragma unroll
            for (int i = 0; i < 16; ++i) {
                const int e = i * 32 + lane;
                float v = al0 * sCg[e] + al1 * sCg[512 + e] + al2 * sCg[1024 + e] + al3 * sCg[1536 + e];
                sCt[i * 33 + lane] = v;
                __builtin_nontemporal_store(v, oC + e);
            }
        }
        __syncthreads();

        // ---- mu_t = A_t @ mu + B_t @ u ----
        float mn = 0.f;
#pragma unroll
        for (int j = 0; j < 32; ++j) mn = fmaf(sAt[lane * 33 + j], __shfl(mur, j), mn);
#pragma unroll
        for (int j = 0; j < 8; ++j)  mn = fmaf(sBt[lane * 9 + j], uv[j], mn);

        // ---- GEMM1: T = A_t * Sigma  (v_wmma_f32_16x16x4_f32, 2x2 tiles, K=32) ----
#pragma unroll
        for (int mi = 0; mi < 2; ++mi)
#pragma unroll
        for (int ni = 0; ni < 2; ++ni) {
            v8f acc = {0.f, 0.f, 0.f, 0.f, 0.f, 0.f, 0.f, 0.f};
#pragma unroll
            for (int kc = 0; kc < 32; kc += 4) {
                const int kb = kc + 2 * hi4;
                v2f a, bb;
                a.x  = sAt[(mi * 16 + lo) * 33 + kb];
                a.y  = sAt[(mi * 16 + lo) * 33 + kb + 1];
                bb.x = sS[kb * 33 + ni * 16 + lo];
                bb.y = sS[(kb + 1) * 33 + ni * 16 + lo];
                acc = __builtin_amdgcn_wmma_f32_16x16x4_f32(false, a, false, bb, (short)0, acc, false, false);
            }
#pragma unroll
            for (int rr = 0; rr < 8; ++rr)
                sT[(mi * 16 + rr + 8 * hi4) * 33 + ni * 16 + lo] = acc[rr];
        }
        __syncthreads();

        // ---- GEMM2: Sigma_new = T * A_t^T + Q ----
        {
            float* oS = out + OFF_SIG + bt * (DZ * DZ);
#pragma unroll
            for (int mi = 0; mi < 2; ++mi)
#pragma unroll
            for (int ni = 0; ni < 2; ++ni) {
                v8f acc = {0.f, 0.f, 0.f, 0.f, 0.f, 0.f, 0.f, 0.f};
#pragma unroll
                for (int kc = 0; kc < 32; kc += 4) {
                    const int kb = kc + 2 * hi4;
                    v2f a, bb;
                    a.x  = sT[(mi * 16 + lo) * 33 + kb];
                    a.y  = sT[(mi * 16 + lo) * 33 + kb + 1];
                    bb.x = sAt[(ni * 16 + lo) * 33 + kb];      // A^T[k][n] = A[n][k]
                    bb.y = sAt[(ni * 16 + lo) * 33 + kb + 1];
                    acc = __builtin_amdgcn_wmma_f32_16x16x4_f32(false, a, false, bb, (short)0, acc, false, false);
                }
#pragma unroll
                for (int rr = 0; rr < 8; ++rr) {
                    const int row = mi * 16 + rr + 8 * hi4, col = ni * 16 + lo;
                    const float v = acc[rr] + sQ[row * 33 + col];
                    sS[row * 33 + col] = v;
                    __builtin_nontemporal_store(v, oS + row * 32 + col);
                }
            }
        }
        __syncthreads();

        // ---- Cholesky, fully in registers: lane i owns row i ----
        float rw[32];
#pragma unroll
        for (int k = 0; k < 32; ++k) rw[k] = sS[lane * 33 + k];
#pragma unroll
        for (int j = 0; j < 32; ++j) {
            const float d   = __shfl(rw[j], j);                 // pivot
            const float sj  = sqrtf(d);
            const float inv = __builtin_amdgcn_rcpf(sj);
            const float cj  = (lane > j) ? rw[j] * inv : ((lane == j) ? sj : 0.f);
            rw[j] = cj;
#pragma unroll
            for (int k2 = j + 1; k2 < 32; ++k2)
                rw[k2] = fmaf(-cj, __shfl(cj, k2), rw[k2]);      // rank-1 update
        }

        // ---- z = mu + L @ eps ----
        float zi = mn;
#pragma unroll
        for (int j = 0; j < 32; ++j) {
            const float lij = (j <= lane) ? rw[j] : 0.f;
            zi = fmaf(lij, __shfl(epsr, j), zi);
        }
        zv[lane] = zi;
        __builtin_nontemporal_store(zi, out + OFF_Z + bt * DZ + lane);
        __builtin_nontemporal_store(mn, out + OFF_MU + bt * DZ + lane);
        mur = mn;
        __syncthreads();

        // ---- a = C_t @ z ----
        if (lane < DA) {
            float av = 0.f;
#pragma unroll
            for (int j = 0; j < 32; ++j) av = fmaf(sCt[lane * 33 + j], zv[j], av);
            __builtin_nontemporal_store(av, out + OFF_A + bt * DA + lane);
        }

        // ---- GRU cell: lane owns gate columns n = lane + 32*jj ----
        float gx[6], gh[6];
#pragma unroll
        for (int jj = 0; jj < 6; ++jj) { gx[jj] = bxp[lane + 32 * jj]; gh[jj] = bhp[lane + 32 * jj]; }
#pragma unroll 4
        for (int k = 0; k < 32; ++k) {
            const float zk = zv[k];
            const float* w = Wx + k * 192 + lane;
#pragma unroll
            for (int jj = 0; jj < 6; ++jj) gx[jj] = fmaf(w[32 * jj], zk, gx[jj]);
        }
#pragma unroll 4
        for (int k = 0; k < 64; ++k) {
            const float hk = hv[k];
            const float* w = Wh + k * 192 + lane;
#pragma unroll
            for (int jj = 0; jj < 6; ++jj) gh[jj] = fmaf(w[32 * jj], hk, gh[jj]);
        }
        const float r0  = sigm(gx[0] + gh[0]);
        const float r1  = sigm(gx[1] + gh[1]);
        const float zg0 = sigm(gx[2] + gh[2]);
        const float zg1 = sigm(gx[3] + gh[3]);
        const float n0  = tanhf(fmaf(r0, gh[4], gx[4]));
        const float n1  = tanhf(fmaf(r1, gh[5], gx[5]));
        const float hn0 = fmaf(zg0, hr0 - n0, n0);   // (1-z)*n + z*h
        const float hn1 = fmaf(zg1, hr1 - n1, n1);
        __syncthreads();
        hv[lane] = hn0; hv[lane + 32] = hn1; hr0 = hn0; hr1 = hn1;
        __syncthreads();

        // ---- alpha_new = softmax(h_new @ Wo + bo) on lanes 0..3 ----
        if (lane < KMIX) {
            float lg = bop[lane];
#pragma unroll 8
            for (int k = 0; k < 64; ++k) lg = fmaf(hv[k], Wo[k * 4 + lane], lg);
            float m = lg;
            m = fmaxf(m, __shfl_xor(m, 1, 4));
            m = fmaxf(m, __shfl_xor(m, 2, 4));
            const float e = __expf(lg - m);
            float s = e;
            s += __shfl_xor(s, 1, 4);
            s += __shfl_xor(s, 2, 4);
            salpha[lane] = e / s;
        }
        __syncthreads();
    }
}

extern "C" void kernel_launch(void* const* d_in, const int* in_sizes, int n_in,
                              void* d_out, int out_size, void* d_ws, size_t ws_size,
                              hipStream_t stream) {
    lgssm_rollout<<<dim3(BATCH), dim3(32), 0, stream>>>(
        (const float*)d_in[0],  (const float*)d_in[1],  (const float*)d_in[2],
        (const float*)d_in[3],  (const float*)d_in[4],  (const float*)d_in[5],
        (const float*)d_in[6],  (const float*)d_in[7],  (const float*)d_in[8],
        (const float*)d_in[9],  (const float*)d_in[10], (const float*)d_in[11],
        (const float*)d_in[12], (const float*)d_in[13], (const float*)d_in[14],
        (const float*)d_in[15], (float*)d_out);
}